// EyeWaveWithPostProcess_70531952935539
// MI455X (gfx1250) — compile-verified
//
#include <hip/hip_runtime.h>

#define BATCH   32
#define CH      7          // 5 + NC
#define HW      80
#define NCELL   6400       // 80*80
#define NPAD    8192       // pow2 pad for bitonic sort
#define KTOP    1024       // padded PRE_NMS
#define PRE     1000
#define MAXDET  300
#define CONF_T  0.25f
#define IOU_T   0.45f
#define STRIDEF 8.0f
#define MWORDS  32         // KTOP / 32 bit-words per mask row

typedef __attribute__((ext_vector_type(2))) float v2f;
typedef __attribute__((ext_vector_type(8))) float v8f;
typedef __attribute__((ext_vector_type(4))) unsigned u32x4;
typedef __attribute__((ext_vector_type(8))) int i32x8;
typedef __attribute__((ext_vector_type(4))) int i32x4;

#if defined(__has_builtin)
#if __has_builtin(__builtin_amdgcn_tensor_load_to_lds) && \
    __has_builtin(__builtin_amdgcn_s_wait_tensorcnt)
#define HAVE_TDM 1
#endif
#endif

__device__ __forceinline__ float sigmoidf(float x) {
  return 1.0f / (1.0f + __expf(-x));
}

// ---------------------------------------------------------------------------
// Kernel 1: grid decode + score/threshold.  Coalesced channel-strided reads.
// ---------------------------------------------------------------------------
__global__ void k_decode(const float* __restrict__ raw,
                         float* __restrict__ score,
                         float* __restrict__ boxes,
                         float* __restrict__ cls) {
  int gid  = blockIdx.x * blockDim.x + threadIdx.x;
  int b    = gid >> 13;            // / NPAD
  int cell = gid & (NPAD - 1);
  if (b >= BATCH) return;
  float* sp = score + (size_t)b * NPAD + cell;
  if (cell >= NCELL) { *sp = -1.0f; return; }   // pad sorts below real zeros

  const float* p = raw + (size_t)b * (CH * NCELL) + cell;
  __builtin_prefetch(p, 0, 1);                  // global_prefetch_b8

  float t0 = p[0 * NCELL], t1 = p[1 * NCELL];
  float t2 = p[2 * NCELL], t3 = p[3 * NCELL];
  float t4 = p[4 * NCELL], t5 = p[5 * NCELL], t6 = p[6 * NCELL];

  float gx = (float)(cell % HW);
  float gy = (float)(cell / HW);
  float cx = (sigmoidf(t0) + gx) * STRIDEF;
  float cy = (sigmoidf(t1) + gy) * STRIDEF;
  float w  = __expf(t2) * STRIDEF;
  float h  = __expf(t3) * STRIDEF;
  float obj = sigmoidf(t4);
  float c0 = sigmoidf(t5), c1 = sigmoidf(t6);
  float cconf = fmaxf(c0, c1);
  float cid   = (c1 > c0) ? 1.0f : 0.0f;        // argmax, first-wins ties
  float s = obj * cconf;
  s = (s > CONF_T) ? s : 0.0f;

  *sp = s;
  float* bp = boxes + ((size_t)b * NCELL + cell) * 4;
  bp[0] = cx; bp[1] = cy; bp[2] = w; bp[3] = h;
  cls[(size_t)b * NCELL + cell] = cid;
}

// ---------------------------------------------------------------------------
// In-LDS bitonic sort, descending, key+payload.
// ---------------------------------------------------------------------------
template <typename IT>
__device__ __forceinline__ void bitonic_desc(float* s, IT* idx, int n, int tid, int nthr) {
  for (int k = 2; k <= n; k <<= 1) {
    for (int j = k >> 1; j > 0; j >>= 1) {
      for (int i = tid; i < n; i += nthr) {
        int ixj = i ^ j;
        if (ixj > i) {
          bool up = (i & k) == 0;
          float si = s[i], sj = s[ixj];
          bool sw = up ? (si < sj) : (si > sj);
          if (sw) {
            s[i] = sj; s[ixj] = si;
            IT t = idx[i]; idx[i] = idx[ixj]; idx[ixj] = t;
          }
        }
      }
      __syncthreads();
    }
  }
}

// ---------------------------------------------------------------------------
// Kernel 2: per-image top-1024 (sorted) of 8192 padded scores.  48 KB LDS.
// Pad entries (k >= PRE) become zero-area sentinel boxes at -1e9 -> IoU 0.
// ---------------------------------------------------------------------------
__global__ void __launch_bounds__(1024) k_topk(const float* __restrict__ score,
                                               const float* __restrict__ boxes,
                                               const float* __restrict__ cls,
                                               float* __restrict__ sk,
                                               float* __restrict__ bk,
                                               float* __restrict__ ck) {
  __shared__ float          s_s[NPAD];   // 32 KB
  __shared__ unsigned short s_i[NPAD];   // 16 KB
  int b = blockIdx.x, tid = threadIdx.x;
  for (int i = tid; i < NPAD; i += 1024) {
    s_s[i] = score[(size_t)b * NPAD + i];
    s_i[i] = (unsigned short)i;
  }
  __syncthreads();
  bitonic_desc(s_s, s_i, NPAD, tid, 1024);

  float s  = s_s[tid];
  int   id = (int)s_i[tid];
  float o0, o1, o2, o3, oc, os;
  if (tid < PRE && s >= 0.0f) {
    const float* bp = boxes + ((size_t)b * NCELL + id) * 4;
    o0 = bp[0]; o1 = bp[1]; o2 = bp[2]; o3 = bp[3];
    oc = cls[(size_t)b * NCELL + id];
    os = s;
  } else {
    o0 = -1.0e9f; o1 = -1.0e9f; o2 = 0.0f; o3 = 0.0f; oc = 0.0f; os = 0.0f;
  }
  sk[(size_t)b * KTOP + tid] = os;
  float* bo = bk + ((size_t)b * KTOP + tid) * 4;
  bo[0] = o0; bo[1] = o1; bo[2] = o2; bo[3] = o3;
  ck[(size_t)b * KTOP + tid] = oc;
}

// ---------------------------------------------------------------------------
// Kernel 3: 1024x1024 suppression bitmask, 16x16 tiles.
//  * 16 KB box tile staged global->LDS by the Tensor Data Mover (1-D D#,
//    4096 x f32 contiguous), completion via s_wait_tensorcnt.
//  * union term (area_i + area_j) on the matrix pipe: rank-2 outer sum via
//    V_WMMA_F32_16X16X4_F32 (A[m]=[area_i,1,0,0], B[:,n]=[1,area_j,0,0]).
//  * intersection on VALU; iou>T tested division-free as inter > T*union.
//  * bits packed into LDS with ds_or atomics.
// ---------------------------------------------------------------------------
__global__ void __launch_bounds__(256) k_iou_mask(const float* __restrict__ bk,
                                                  unsigned* __restrict__ mask) {
  __shared__ float s_jbox[KTOP * 4];                       // 16 KB raw cxcywh
  __shared__ float s_jx1[KTOP], s_jy1[KTOP], s_jx2[KTOP], s_jy2[KTOP], s_ja[KTOP];
  __shared__ float s_ix1[16], s_iy1[16], s_ix2[16], s_iy2[16], s_ia[16];
  __shared__ unsigned s_m[16 * MWORDS];

  int b  = blockIdx.y;
  int ti = blockIdx.x;         // i-row tile (16 rows)
  int tid = threadIdx.x;

  const float* gsrc = bk + (size_t)b * KTOP * 4;

#ifdef HAVE_TDM
  if (tid == 0) {
    // ---- Tensor DMA descriptor (D#): 1-D tensor, 4096 x f32, contiguous ----
    unsigned long long ga = (unsigned long long)(uintptr_t)gsrc;
    unsigned lds_addr = (unsigned)(uintptr_t)(void*)s_jbox;   // LDS byte offset
    u32x4 g0;
    g0.x = 1u;                                           // count=1, is_restore=0
    g0.y = lds_addr;                                     // bits [63:32]
    g0.z = (unsigned)(ga & 0xFFFFFFFFu);                 // global_addr lo
    g0.w = ((unsigned)(ga >> 32) & 0x01FFFFFFu) | (2u << 30);  // addr hi | type=2
    i32x8 g1;
    g1[0] = 0x00020000;                 // data_size=2 (4 bytes); mask/flags 0
    g1[1] = (int)(4096u << 16);         // tensor_dim0[15:0] = 4096
    g1[2] = (int)(1u << 16);            // tensor_dim0 hi=0, tensor_dim1=1
    g1[3] = (int)(4096u << 16);         // tensor_dim1 hi=0, tile_dim0=4096
    g1[4] = 0;                          // tile_dim1=0, tile_dim2=0 (unused)
    g1[5] = 4096;                       // tensor_dim0_stride lo
    g1[6] = 0;                          // stride hi, tensor_dim1_stride lo
    g1[7] = 0;
    i32x4 gz4 = {0, 0, 0, 0};
    i32x8 gz8 = {0, 0, 0, 0, 0, 0, 0, 0};
    __builtin_amdgcn_tensor_load_to_lds(g0, g1, gz4, gz4, gz8, 0);  // TENSORcnt++
    __builtin_amdgcn_s_wait_tensorcnt(0);                           // s_wait_tensorcnt
  }
#else
  for (int j = tid; j < KTOP * 4; j += 256) s_jbox[j] = gsrc[j];
#endif
  __syncthreads();

  for (int j = tid; j < KTOP; j += 256) {
    float cx = s_jbox[j * 4 + 0], cy = s_jbox[j * 4 + 1];
    float w  = s_jbox[j * 4 + 2], h  = s_jbox[j * 4 + 3];
    float x1 = cx - 0.5f * w, x2 = cx + 0.5f * w;
    float y1 = cy - 0.5f * h, y2 = cy + 0.5f * h;
    s_jx1[j] = x1; s_jy1[j] = y1; s_jx2[j] = x2; s_jy2[j] = y2;
    s_ja[j] = (x2 - x1) * (y2 - y1);
  }
  if (tid < 16) {
    int i = ti * 16 + tid;
    float cx = s_jbox[i * 4 + 0], cy = s_jbox[i * 4 + 1];
    float w  = s_jbox[i * 4 + 2], h  = s_jbox[i * 4 + 3];
    float x1 = cx - 0.5f * w, x2 = cx + 0.5f * w;
    float y1 = cy - 0.5f * h, y2 = cy + 0.5f * h;
    s_ix1[tid] = x1; s_iy1[tid] = y1; s_ix2[tid] = x2; s_iy2[tid] = y2;
    s_ia[tid] = (x2 - x1) * (y2 - y1);
  }
  for (int wds = tid; wds < 16 * MWORDS; wds += 256) s_m[wds] = 0u;
  __syncthreads();

  int wave = tid >> 5, lane = tid & 31;
  bool lo = lane < 16;
  int n  = lane & 15;
  int hi = lane >> 4;
  // A-matrix 16x4: lanes 0-15 carry K=0,1 for row m=lane; lanes 16-31 (K=2,3) zero
  v2f A;
  A.x = lo ? s_ia[n] : 0.0f;
  A.y = lo ? 1.0f : 0.0f;

  for (int tj = wave; tj < KTOP / 16; tj += 8) {
    int j0 = tj * 16;
    // B-matrix 4x16: K=0 row = 1, K=1 row = area_j[n]; K=2,3 zero
    v2f Bv;
    Bv.x = lo ? 1.0f : 0.0f;
    Bv.y = lo ? s_ja[j0 + n] : 0.0f;
    v8f acc = {0.f, 0.f, 0.f, 0.f, 0.f, 0.f, 0.f, 0.f};
    // D[m][n] = area_i[m] + area_j[n]
    acc = __builtin_amdgcn_wmma_f32_16x16x4_f32(false, A, false, Bv,
                                                (short)0, acc, false, false);

    float jx1 = s_jx1[j0 + n], jy1 = s_jy1[j0 + n];
    float jx2 = s_jx2[j0 + n], jy2 = s_jy2[j0 + n];
    int col = j0 + n;
    unsigned bit = 1u << (col & 31);
    int word = col >> 5;
#pragma unroll
    for (int r = 0; r < 8; ++r) {
      int m = r + (hi << 3);          // C/D layout: VGPR r -> row r (lo) / r+8 (hi)
      float iw = fminf(s_ix2[m], jx2) - fmaxf(s_ix1[m], jx1);
      float ih = fminf(s_iy2[m], jy2) - fmaxf(s_iy1[m], jy1);
      iw = fmaxf(iw, 0.0f); ih = fmaxf(ih, 0.0f);
      float inter = iw * ih;
      float uni = acc[r] - inter + 1e-9f;   // area_i + area_j - inter + eps
      if (inter > IOU_T * uni)
        atomicOr(&s_m[m * MWORDS + word], bit);   // ds_or_b32
    }
  }
  __syncthreads();
  for (int wds = tid; wds < 16 * MWORDS; wds += 256) {
    int row = wds >> 5, wd = wds & 31;
    mask[((size_t)b * KTOP + ti * 16 + row) * MWORDS + wd] = s_m[wds];
  }
}

// ---------------------------------------------------------------------------
// Kernel 4: serial greedy scan (one wave32: lane w owns keep-word w, 32x32 =
// 1024 columns), then top-300 of kept scores + row emission.
// ---------------------------------------------------------------------------
__global__ void __launch_bounds__(1024) k_finalize(const float* __restrict__ sk,
                                                   const float* __restrict__ bk,
                                                   const float* __restrict__ ck,
                                                   const unsigned* __restrict__ mask,
                                                   float* __restrict__ out) {
  __shared__ float s_s[KTOP];
  __shared__ int   s_i[KTOP];
  __shared__ unsigned s_keep[MWORDS];
  int b = blockIdx.x, tid = threadIdx.x;

  if (tid < 32) {
    unsigned keepw = 0xFFFFFFFFu;
    const unsigned* mrow = mask + (size_t)b * KTOP * MWORDS;
    for (int i = 0; i < PRE; ++i) {
      int wi = i >> 5, bi = i & 31;
      unsigned kw_i = __shfl(keepw, wi, 32);
      if ((kw_i >> bi) & 1u) {                 // row i still kept -> suppress
        unsigned row = mrow[(size_t)i * MWORDS + tid];
        if (tid == wi)      row &= ~((2u << bi) - 1u);  // only j > i
        else if (tid < wi)  row = 0u;
        keepw &= ~row;
      }
    }
    s_keep[tid] = keepw;
  }
  __syncthreads();

  bool kept = (s_keep[tid >> 5] >> (tid & 31)) & 1u;
  float s = sk[(size_t)b * KTOP + tid];
  s_s[tid] = kept ? s : 0.0f;
  s_i[tid] = tid;
  __syncthreads();
  bitonic_desc(s_s, s_i, KTOP, tid, 1024);

  if (tid < MAXDET) {
    float fd = s_s[tid];
    int fi = s_i[tid];
    float r0 = 0.f, r1 = 0.f, r2 = 0.f, r3 = 0.f, r4 = 0.f, r5 = 0.f;
    if (fd > 0.0f) {
      const float* bp = bk + ((size_t)b * KTOP + fi) * 4;
      r0 = bp[0]; r1 = bp[1]; r2 = bp[2]; r3 = bp[3];
      r4 = fd;
      r5 = ck[(size_t)b * KTOP + fi];
    }
    float* o = out + ((size_t)b * MAXDET + tid) * 6;
    o[0] = r0; o[1] = r1; o[2] = r2; o[3] = r3; o[4] = r4; o[5] = r5;
  }
}

// ---------------------------------------------------------------------------
extern "C" void kernel_launch(void* const* d_in, const int* in_sizes, int n_in,
                              void* d_out, int out_size, void* d_ws, size_t ws_size,
                              hipStream_t stream) {
  (void)in_sizes; (void)n_in; (void)out_size; (void)ws_size;
  const float* raw = (const float*)d_in[0];
  float* out = (float*)d_out;
  char* ws = (char*)d_ws;

  size_t off = 0;
  float* score = (float*)(ws + off); off += (size_t)BATCH * NPAD * sizeof(float);        // 1 MB
  float* boxes = (float*)(ws + off); off += (size_t)BATCH * NCELL * 4 * sizeof(float);   // 3.3 MB
  float* cls   = (float*)(ws + off); off += (size_t)BATCH * NCELL * sizeof(float);       // 0.8 MB
  float* sk    = (float*)(ws + off); off += (size_t)BATCH * KTOP * sizeof(float);
  float* bk    = (float*)(ws + off); off += (size_t)BATCH * KTOP * 4 * sizeof(float);
  float* ck    = (float*)(ws + off); off += (size_t)BATCH * KTOP * sizeof(float);
  unsigned* mask = (unsigned*)(ws + off); off += (size_t)BATCH * KTOP * MWORDS * sizeof(unsigned); // 4 MB

  k_decode<<<(BATCH * NPAD) / 256, 256, 0, stream>>>(raw, score, boxes, cls);
  k_topk<<<BATCH, 1024, 0, stream>>>(score, boxes, cls, sk, bk, ck);
  dim3 g3(KTOP / 16, BATCH);
  k_iou_mask<<<g3, 256, 0, stream>>>(bk, mask);
  k_finalize<<<BATCH, 1024, 0, stream>>>(sk, bk, ck, mask, out);
}